// MyModel_87522843559525
// MI455X (gfx1250) — compile-verified
//
#include <hip/hip_runtime.h>
#include <hip/hip_bf16.h>

typedef __attribute__((ext_vector_type(2))) float v2f;
typedef __attribute__((ext_vector_type(8))) float v8f;

#define B_ 128
#define T_ 1024
#define F_ 256
#define U_ 100
#define G_ 400   // 4*U

static __device__ __forceinline__ v8f wmma_f32(v2f a, v2f b, v8f c) {
    // D = A(16x4,f32) * B(4x16,f32) + C(16x16,f32)
    return __builtin_amdgcn_wmma_f32_16x16x4_f32(
        /*neg_a=*/false, a, /*neg_b=*/false, b,
        /*c_mod=*/(short)0, c, /*reuse_a=*/false, /*reuse_b=*/false);
}

static __device__ __forceinline__ float sigmoidf_(float x) {
    return 1.0f / (1.0f + __expf(-x));
}

// ---------------------------------------------------------------------------
// Kernel 1: xz[(b*T+t), g] = sum_f x[(b*T+t), f] * kernel[f, g] + bias[g]
// M = B*T = 131072, K = F = 256, N = G = 400 (exactly 25 tiles of 16).
// Block = 256 threads = 8 waves; each wave owns one 16x16 output tile,
// block covers 8 consecutive M-tiles sharing one kernel N-column tile in LDS.
// ---------------------------------------------------------------------------
__global__ __launch_bounds__(256) void xz_gemm(const float* __restrict__ x,
                                               const float* __restrict__ kern,
                                               const float* __restrict__ bias,
                                               float* __restrict__ xz)
{
    __shared__ float Kt[F_ * 16];   // 256 x 16 f32 tile of `kernel` = 16 KB

    const int tid = threadIdx.x;
    const int n0  = blockIdx.y * 16;

    // Stage kernel tile: thread k loads row k, 16 floats (float4 x 4).
    {
        const float* src = kern + (size_t)tid * G_ + n0;   // 16B aligned
        #pragma unroll
        for (int j = 0; j < 16; j += 4) {
            float4 v = *(const float4*)(src + j);
            Kt[tid * 16 + j + 0] = v.x;
            Kt[tid * 16 + j + 1] = v.y;
            Kt[tid * 16 + j + 2] = v.z;
            Kt[tid * 16 + j + 3] = v.w;
        }
    }
    __syncthreads();

    const int wave = tid >> 5;
    const int lane = tid & 31;
    const int row  = lane & 15;     // A-row / B-col / D-col index
    const int hi   = lane >> 4;     // lane half
    const int kb   = hi * 2;        // K sub-offset for A/B frags
    const int mb   = hi * 8;        // M block for C/D frags

    const size_t mtile = (size_t)blockIdx.x * 8 + wave;
    const float* xrow  = x + (mtile * 16 + row) * F_;

    v8f acc = {};
    for (int k0 = 0; k0 < F_; k0 += 4) {
        v2f a = *(const v2f*)(xrow + k0 + kb);            // global_load_b64
        v2f b;
        b.x = Kt[(k0 + kb)     * 16 + row];
        b.y = Kt[(k0 + kb + 1) * 16 + row];
        acc = wmma_f32(a, b, acc);
    }

    const float bv     = bias[n0 + row];
    const size_t mrow0 = mtile * 16 + mb;
    #pragma unroll
    for (int r = 0; r < 8; ++r) {
        xz[(mrow0 + r) * G_ + n0 + row] = acc[r] + bv;
    }
}

// ---------------------------------------------------------------------------
// Kernel 2: masked LSTM scan + final dense head.
// 8 blocks x 256 threads; block handles 16 batch rows (independent sequences).
// Wrec (100x400 f32 = 160 KB) is staged ONCE into dynamic LDS (CDNA5 allows
// up to 320 KB per workgroup), so every per-step WMMA B-frag is an LDS hit
// instead of an L2 round-trip on the serial critical path.
// Per step: z(16x400) = h(16x100) @ Wrec(100x400) + xz[:, t, :]  via WMMA,
// then gate math in LDS, masked state update, 1024 iterations.
// ---------------------------------------------------------------------------
__global__ __launch_bounds__(256) void lstm_scan(const float* __restrict__ xz,
                                                 const float* __restrict__ Wr,
                                                 const int*   __restrict__ lengths,
                                                 const float* __restrict__ dk,
                                                 const float* __restrict__ db,
                                                 float* __restrict__ out)
{
    extern __shared__ float WrL[];   // 100*400 f32 = 160 KB (dynamic LDS)
    __shared__ float hL[16 * U_];    // 6.4 KB
    __shared__ float cL[16 * U_];    // 6.4 KB
    __shared__ float zL[16 * G_];    // 25.6 KB
    __shared__ int   lenL[16];

    const int tid = threadIdx.x;
    const int b0  = blockIdx.x * 16;

    // Stage Wrec into LDS (float4-wide, coalesced): 10000 x b128.
    for (int i = tid; i < (U_ * G_) / 4; i += 256) {
        ((float4*)WrL)[i] = ((const float4*)Wr)[i];
    }
    for (int i = tid; i < 16 * U_; i += 256) { hL[i] = 0.0f; cL[i] = 0.0f; }
    if (tid < 16) lenL[tid] = lengths[b0 + tid];
    __syncthreads();

    const int wave = tid >> 5;
    const int lane = tid & 31;
    const int row  = lane & 15;
    const int hi   = lane >> 4;
    const int kb   = hi * 2;
    const int mb   = hi * 8;

    for (int t = 0; t < T_; ++t) {
        // --- z = h @ Wrec (25 N-tiles split across 8 waves) + xz slice ---
        for (int nt = wave; nt < 25; nt += 8) {
            const int n0 = nt * 16;
            v8f acc = {};
            for (int k0 = 0; k0 < U_; k0 += 4) {
                v2f a;
                a.x = hL[row * U_ + k0 + kb];
                a.y = hL[row * U_ + k0 + kb + 1];
                v2f b;   // bank-conflict free: rows differ by 800 words = +32 banks
                b.x = WrL[(k0 + kb)     * G_ + n0 + row];
                b.y = WrL[(k0 + kb + 1) * G_ + n0 + row];
                acc = wmma_f32(a, b, acc);
            }
            if (t + 1 < T_) {   // prefetch next step's xz slice (global_prefetch_b8)
                __builtin_prefetch(&xz[(((size_t)(b0 + mb)) * T_ + (t + 1)) * G_ + n0 + row], 0, 0);
            }
            #pragma unroll
            for (int r = 0; r < 8; ++r) {
                const int m = mb + r;
                const size_t off = (((size_t)(b0 + m)) * T_ + t) * G_ + n0 + row;
                zL[m * G_ + n0 + row] = acc[r] + xz[off];
            }
        }
        __syncthreads();

        // --- gates + masked state update (1600 elements / 256 threads) ---
        for (int i = tid; i < 16 * U_; i += 256) {
            const int m = i / U_;
            const int u = i - m * U_;
            const float zi = zL[m * G_ + u];
            const float zf = zL[m * G_ + U_ + u];
            const float zg = zL[m * G_ + 2 * U_ + u];
            const float zo = zL[m * G_ + 3 * U_ + u];
            const float ig = sigmoidf_(zi);
            const float fg = sigmoidf_(zf);
            const float gg = tanhf(zg);
            const float og = sigmoidf_(zo);
            const float cn = fg * cL[i] + ig * gg;
            const float hn = og * tanhf(cn);
            if (t < lenL[m]) { cL[i] = cn; hL[i] = hn; }
        }
        __syncthreads();
    }

    // --- final dense head: out[b] = sigmoid(h[b,:] @ dk + db) ---
    if (tid < 16) {
        float s = db[0];
        for (int u = 0; u < U_; ++u) s += hL[tid * U_ + u] * dk[u];
        out[b0 + tid] = sigmoidf_(s);
    }
}

extern "C" void kernel_launch(void* const* d_in, const int* in_sizes, int n_in,
                              void* d_out, int out_size, void* d_ws, size_t ws_size,
                              hipStream_t stream) {
    (void)in_sizes; (void)n_in; (void)out_size; (void)ws_size;
    const float* x       = (const float*)d_in[0];
    const int*   lengths = (const int*)  d_in[1];
    const float* kern    = (const float*)d_in[2];
    const float* Wr      = (const float*)d_in[3];
    const float* bias    = (const float*)d_in[4];
    const float* dk      = (const float*)d_in[5];
    const float* db      = (const float*)d_in[6];
    float*       out     = (float*)d_out;
    float*       xz      = (float*)d_ws;   // (B*T) x 400 f32 ≈ 210 MB scratch

    // Kernel 1: 8192 M-tiles / 8 per block = 1024 blocks, 25 N-tiles.
    dim3 g1((B_ * T_ / 16) / 8, G_ / 16);
    xz_gemm<<<g1, 256, 0, stream>>>(x, kern, bias, xz);

    // Kernel 2: 8 blocks of 16 batch rows each; 160 KB dynamic LDS for Wrec.
    const size_t wrec_lds = (size_t)U_ * G_ * sizeof(float);   // 160000 B
    (void)hipFuncSetAttribute((const void*)lstm_scan,
                              hipFuncAttributeMaxDynamicSharedMemorySize,
                              (int)wrec_lds);
    lstm_scan<<<dim3(B_ / 16), 256, wrec_lds, stream>>>(xz, Wr, lengths, dk, db, out);
}